// PlainGAT_39805756900162
// MI455X (gfx1250) — compile-verified
//
#include <hip/hip_runtime.h>
#include <hip/hip_bf16.h>

typedef __attribute__((ext_vector_type(2))) float v2f;
typedef __attribute__((ext_vector_type(8))) float v8f;

#define Bn   8
#define Nn   512
#define Dn   512
#define Hn   8
#define DVn  64

// ---------------------------------------------------------------------------
// Kernel 1: value projection  value[b,h,n,d] = (inp @ W_value)[b*N+n, h*64+d] + bias
// 64x64 block tile, K-step 16, 8 waves; each wave: two 16x16 f32 WMMA accumulators.
// ---------------------------------------------------------------------------
__global__ __launch_bounds__(256)
void k_value_gemm(const float* __restrict__ inp, const float* __restrict__ W,
                  const float* __restrict__ bias, float* __restrict__ value)
{
    __shared__ float As[64][17];   // M x K tile (padded: conflict-free frag reads)
    __shared__ float Bs[16][68];   // K x N tile (padded)

    const int t    = threadIdx.x;
    const int lane = t & 31, wave = t >> 5;
    const int half = lane >> 4, l15 = lane & 15;
    const int bm   = blockIdx.x * 64;          // global M tile (rows of B*N)
    const int bn   = blockIdx.y * 64;          // global N tile (cols of H*dv)
    const int m0   = (wave & 3) * 16;          // wave's M sub-tile
    const int n0   = (wave >> 2) * 32;         // wave's N sub-tile base (2 frags)

    const int ar = t >> 2, ac = (t & 3) * 4;   // coop-load coords for A (64x16)
    const int br = t >> 4, bc = (t & 15) * 4;  // coop-load coords for B (16x64)

    v8f acc0 = {}; v8f acc1 = {};

    for (int k0 = 0; k0 < Dn; k0 += 16) {
        float4 av = *(const float4*)(inp + (size_t)(bm + ar) * Dn + k0 + ac);
        float4 bv = *(const float4*)(W   + (size_t)(k0 + br) * Dn + bn + bc);
        __syncthreads();                       // prior iteration done reading LDS
        As[ar][ac + 0] = av.x; As[ar][ac + 1] = av.y;
        As[ar][ac + 2] = av.z; As[ar][ac + 3] = av.w;
        Bs[br][bc + 0] = bv.x; Bs[br][bc + 1] = bv.y;
        Bs[br][bc + 2] = bv.z; Bs[br][bc + 3] = bv.w;
        __syncthreads();
#pragma unroll
        for (int kk = 0; kk < 16; kk += 4) {
            v2f a, b0, b1;
            a[0]  = As[m0 + l15][kk + 2 * half + 0];
            a[1]  = As[m0 + l15][kk + 2 * half + 1];
            b0[0] = Bs[kk + 2 * half + 0][n0 + l15];
            b0[1] = Bs[kk + 2 * half + 1][n0 + l15];
            b1[0] = Bs[kk + 2 * half + 0][n0 + 16 + l15];
            b1[1] = Bs[kk + 2 * half + 1][n0 + 16 + l15];
            acc0 = __builtin_amdgcn_wmma_f32_16x16x4_f32(false, a, false, b0,
                                                         (short)0, acc0, false, false);
            acc1 = __builtin_amdgcn_wmma_f32_16x16x4_f32(false, a, false, b1,
                                                         (short)0, acc1, false, false);
        }
    }

    // Scatter C frags into [B,H,N,dv] layout (+ bias)
#pragma unroll
    for (int v = 0; v < 8; ++v) {
        const int R = bm + m0 + v + 8 * half;      // global row = b*512 + n
        const int b = R >> 9, n = R & 511;
        {
            const int C = bn + n0 + l15;           // global col = h*64 + d
            const int h = C >> 6, d = C & 63;
            value[(((size_t)(b * Hn + h)) * Nn + n) * DVn + d] = acc0[v] + bias[C];
        }
        {
            const int C = bn + n0 + 16 + l15;
            const int h = C >> 6, d = C & 63;
            value[(((size_t)(b * Hn + h)) * Nn + n) * DVn + d] = acc1[v] + bias[C];
        }
    }
}

// ---------------------------------------------------------------------------
// Kernel 2: s_src[b,h,n] = value[b,h,n,:].w_src[h,:]  (and s_tgt).  One wave/row.
// ---------------------------------------------------------------------------
__global__ __launch_bounds__(256)
void k_scores(const float* __restrict__ value, const float* __restrict__ wsrc,
              const float* __restrict__ wtgt, float* __restrict__ s_src,
              float* __restrict__ s_tgt)
{
    const int row  = blockIdx.x * 8 + (threadIdx.x >> 5);  // (b*H+h)*N + n
    const int lane = threadIdx.x & 31;
    const int h    = (row >> 9) & 7;
    const float v0 = value[(size_t)row * DVn + lane];
    const float v1 = value[(size_t)row * DVn + 32 + lane];
    float ss = v0 * wsrc[h * DVn + lane] + v1 * wsrc[h * DVn + 32 + lane];
    float st = v0 * wtgt[h * DVn + lane] + v1 * wtgt[h * DVn + 32 + lane];
#pragma unroll
    for (int o = 16; o > 0; o >>= 1) {
        ss += __shfl_xor(ss, o, 32);
        st += __shfl_xor(st, o, 32);
    }
    if (lane == 0) { s_src[row] = ss; s_tgt[row] = st; }
}

// ---------------------------------------------------------------------------
// Kernel 3: one block per (b,h,i) row: score -> LeakyReLU -> mask -> softmax
// -> degree-normalize -> L1 renorm -> zero invalid. Writes attn into d_out.
// ---------------------------------------------------------------------------
__device__ __forceinline__ float block_reduce(float v, float* red, int t, bool isMax)
{
    red[t] = v;
    __syncthreads();
#pragma unroll
    for (int s = 128; s > 0; s >>= 1) {
        if (t < s) red[t] = isMax ? fmaxf(red[t], red[t + s]) : (red[t] + red[t + s]);
        __syncthreads();
    }
    const float r = red[0];
    __syncthreads();
    return r;
}

__global__ __launch_bounds__(256)
void k_attn(const float* __restrict__ s_src, const float* __restrict__ s_tgt,
            const unsigned char* __restrict__ mask, const int* __restrict__ adj,
            float* __restrict__ attn)
{
    __shared__ float red[256];
    const int row = blockIdx.x;                 // (b*H+h)*N + i
    const int bh  = row >> 9, i = row & 511, b = bh >> 3;
    const int t   = threadIdx.x;
    const float st = s_tgt[row];

    const int j0 = t, j1 = t + 256;
    const int a0 = adj[(size_t)row * Nn + j0];
    const int a1 = adj[(size_t)row * Nn + j1];
    const bool inv0 = (mask[((size_t)b * Nn + i) * Nn + j0] != 0) || (a0 == 0);
    const bool inv1 = (mask[((size_t)b * Nn + i) * Nn + j1] != 0) || (a1 == 0);

    float sc0 = st + s_src[(size_t)bh * Nn + j0];
    float sc1 = st + s_src[(size_t)bh * Nn + j1];
    sc0 = sc0 > 0.0f ? sc0 : 0.2f * sc0;
    sc1 = sc1 > 0.0f ? sc1 : 0.2f * sc1;
    if (inv0) sc0 = -1e12f;
    if (inv1) sc1 = -1e12f;

    const float m = block_reduce(fmaxf(sc0, sc1), red, t, true);
    float e0 = expf(sc0 - m), e1 = expf(sc1 - m);
    const float s = block_reduce(e0 + e1, red, t, false);
    float p0 = e0 / s, p1 = e1 / s;

    p0 *= (a0 != 0) ? (1.0f / (float)a0) : 1e-12f;   // 1/INF
    p1 *= (a1 != 0) ? (1.0f / (float)a1) : 1e-12f;
    const float l1 = block_reduce(fabsf(p0) + fabsf(p1), red, t, false);
    const float rinv = 1.0f / fmaxf(l1, 1e-12f);
    p0 *= rinv; p1 *= rinv;
    if (inv0) p0 = 0.0f;
    if (inv1) p1 = 0.0f;

    attn[(size_t)row * Nn + j0] = p0;
    attn[(size_t)row * Nn + j1] = p1;
}

// ---------------------------------------------------------------------------
// Kernel 4: out[b,h] = attn[b,h] (512x512) @ value[b,h] (512x64), fused with
// final = out(+concat) + inp + final_bias. Same WMMA tiling as kernel 1.
// ---------------------------------------------------------------------------
__global__ __launch_bounds__(256)
void k_out_gemm(const float* __restrict__ attn, const float* __restrict__ value,
                const float* __restrict__ inp, const float* __restrict__ fbias,
                float* __restrict__ final_out)
{
    __shared__ float As[64][17];
    __shared__ float Bs[16][68];

    const int t    = threadIdx.x;
    const int lane = t & 31, wave = t >> 5;
    const int half = lane >> 4, l15 = lane & 15;
    const int bm   = blockIdx.x * 64;       // rows (n) tile
    const int bz   = blockIdx.y;            // b*H + h
    const int m0   = (wave & 3) * 16;
    const int n0   = (wave >> 2) * 32;

    const int ar = t >> 2, ac = (t & 3) * 4;
    const int br = t >> 4, bc = (t & 15) * 4;

    const float* Abase = attn  + (size_t)bz * Nn * Nn;
    const float* Bbase = value + (size_t)bz * Nn * DVn;

    v8f acc0 = {}; v8f acc1 = {};

    for (int k0 = 0; k0 < Nn; k0 += 16) {
        float4 av = *(const float4*)(Abase + (size_t)(bm + ar) * Nn + k0 + ac);
        float4 bv = *(const float4*)(Bbase + (size_t)(k0 + br) * DVn + bc);
        __syncthreads();
        As[ar][ac + 0] = av.x; As[ar][ac + 1] = av.y;
        As[ar][ac + 2] = av.z; As[ar][ac + 3] = av.w;
        Bs[br][bc + 0] = bv.x; Bs[br][bc + 1] = bv.y;
        Bs[br][bc + 2] = bv.z; Bs[br][bc + 3] = bv.w;
        __syncthreads();
#pragma unroll
        for (int kk = 0; kk < 16; kk += 4) {
            v2f a, b0, b1;
            a[0]  = As[m0 + l15][kk + 2 * half + 0];
            a[1]  = As[m0 + l15][kk + 2 * half + 1];
            b0[0] = Bs[kk + 2 * half + 0][n0 + l15];
            b0[1] = Bs[kk + 2 * half + 1][n0 + l15];
            b1[0] = Bs[kk + 2 * half + 0][n0 + 16 + l15];
            b1[1] = Bs[kk + 2 * half + 1][n0 + 16 + l15];
            acc0 = __builtin_amdgcn_wmma_f32_16x16x4_f32(false, a, false, b0,
                                                         (short)0, acc0, false, false);
            acc1 = __builtin_amdgcn_wmma_f32_16x16x4_f32(false, a, false, b1,
                                                         (short)0, acc1, false, false);
        }
    }

    const int b = bz >> 3, h = bz & 7;
#pragma unroll
    for (int v = 0; v < 8; ++v) {
        const int n = bm + m0 + v + 8 * half;
        {
            const int d = n0 + l15;
            const size_t idx = ((size_t)(b * Nn + n)) * Dn + h * DVn + d;
            final_out[idx] = acc0[v] + inp[idx] + fbias[h * DVn + d];
        }
        {
            const int d = n0 + 16 + l15;
            const size_t idx = ((size_t)(b * Nn + n)) * Dn + h * DVn + d;
            final_out[idx] = acc1[v] + inp[idx] + fbias[h * DVn + d];
        }
    }
}

// ---------------------------------------------------------------------------
extern "C" void kernel_launch(void* const* d_in, const int* in_sizes, int n_in,
                              void* d_out, int out_size, void* d_ws, size_t ws_size,
                              hipStream_t stream)
{
    const float*         inp   = (const float*)d_in[0];
    const unsigned char* mask  = (const unsigned char*)d_in[1];   // bool [B,N,N]
    const int*           adj   = (const int*)d_in[2];             // int32 [B,H,N,N]
    const float*         Wv    = (const float*)d_in[3];
    const float*         bv    = (const float*)d_in[4];
    const float*         wsrc  = (const float*)d_in[5];
    const float*         wtgt  = (const float*)d_in[6];
    const float*         fbias = (const float*)d_in[7];

    float* final_out = (float*)d_out;                             // [B,N,D]
    float* attn_out  = final_out + (size_t)Bn * Nn * Dn;          // [B,H,N,N]

    float* value = (float*)d_ws;                                  // [B,H,N,dv] 8 MB
    float* s_src = value + (size_t)Bn * Hn * Nn * DVn;            // [B,H,N]
    float* s_tgt = s_src + (size_t)Bn * Hn * Nn;                  // [B,H,N]

    dim3 g1((Bn * Nn) / 64, Dn / 64);          // 64 x 8
    k_value_gemm<<<g1, 256, 0, stream>>>(inp, Wv, bv, value);

    k_scores<<<(Bn * Hn * Nn) / 8, 256, 0, stream>>>(value, wsrc, wtgt, s_src, s_tgt);

    k_attn<<<Bn * Hn * Nn, 256, 0, stream>>>(s_src, s_tgt, mask, adj, attn_out);

    dim3 g4(Nn / 64, Bn * Hn);                 // 8 x 64
    k_out_gemm<<<g4, 256, 0, stream>>>(attn_out, value, inp, fbias, final_out);
}